// InductiveModel_52441550684206
// MI455X (gfx1250) — compile-verified
//
#include <hip/hip_runtime.h>

typedef float v2f __attribute__((ext_vector_type(2)));
typedef float v8f __attribute__((ext_vector_type(8)));

#define NN 50000
#define FDIM 128
#define BN_EPS 1e-5f

// ---------------- degree count / reciprocal ----------------
__global__ void count_edges(const long long* __restrict__ dst,
                            float* __restrict__ cnt, int E) {
  int i = blockIdx.x * blockDim.x + threadIdx.x;
  if (i < E) atomicAdd(&cnt[(int)dst[i]], 1.0f);
}

__global__ void invert_counts(float* __restrict__ cnt, int n) {
  int i = blockIdx.x * blockDim.x + threadIdx.x;
  if (i < n) cnt[i] = 1.0f / fmaxf(cnt[i], 1.0f);
}

// ---------------- scatter-add aggregation (F=128) ----------------
// one thread per (edge, float4-group); 32 groups of 4 floats per edge
__global__ void scatter_add_f128(const float* __restrict__ X,
                                 const long long* __restrict__ src,
                                 const long long* __restrict__ dst,
                                 float* __restrict__ agg, int E) {
  long long t = (long long)blockIdx.x * blockDim.x + threadIdx.x;
  if (t >= (long long)E * 32) return;
  int e = (int)(t >> 5);
  int q = (int)(t & 31);
  int s = (int)src[e];
  int d = (int)dst[e];
  float4 v = ((const float4*)(X + (long long)s * FDIM))[q];
  float* o = agg + (long long)d * FDIM + q * 4;
  atomicAdd(o + 0, v.x);
  atomicAdd(o + 1, v.y);
  atomicAdd(o + 2, v.z);
  atomicAdd(o + 3, v.w);
}

// ---------------- fused SAGE GEMM:  H = (agg*inv) @ Wl + X @ Wr + b  [BN+ReLU] --
// 16 rows per workgroup, one wave per 16-column tile, f32 WMMA 16x16x4, K=2*128
template <int NOUT, bool BNRELU>
__global__ __launch_bounds__(NOUT * 2) void sage_gemm(
    const float* __restrict__ agg, const float* __restrict__ X,
    const float* __restrict__ inv_cnt,
    const float* __restrict__ Wl, const float* __restrict__ Wr,
    const float* __restrict__ bias,
    const float* __restrict__ gamma, const float* __restrict__ beta,
    const float* __restrict__ mean, const float* __restrict__ var,
    float* __restrict__ H) {
  __shared__ float sA[16][FDIM + 4];  // pad 4: conflict-free column reads
  __shared__ float sX[16][FDIM + 4];

  const int tid  = threadIdx.x;
  const int lane = tid & 31;       // wave32
  const int wave = tid >> 5;
  const int row0 = blockIdx.x * 16;

  // cooperative, coalesced tile load; fold mean-divide into A tile
  for (int idx = tid; idx < 16 * FDIM; idx += NOUT * 2) {
    int r = idx >> 7, c = idx & (FDIM - 1);
    float ic = inv_cnt[row0 + r];
    sA[r][c] = agg[(long long)(row0 + r) * FDIM + c] * ic;
    sX[r][c] = X[(long long)(row0 + r) * FDIM + c];
  }
  __syncthreads();

  const int m    = lane & 15;            // A row / B col within tile
  const int khi  = (lane >> 4) << 1;     // lanes 16-31 hold K+2,K+3
  const int col0 = wave * 16;

  v8f acc = {};
  for (int k = 0; k < FDIM; k += 4) {
    v2f a, b;
    a.x = sA[m][k + khi];
    a.y = sA[m][k + khi + 1];
    b.x = Wl[(k + khi) * NOUT + col0 + m];
    b.y = Wl[(k + khi + 1) * NOUT + col0 + m];
#if __HIP_DEVICE_COMPILE__
    acc = __builtin_amdgcn_wmma_f32_16x16x4_f32(false, a, false, b,
                                                (short)0, acc, false, false);
#endif
  }
  for (int k = 0; k < FDIM; k += 4) {
    v2f a, b;
    a.x = sX[m][k + khi];
    a.y = sX[m][k + khi + 1];
    b.x = Wr[(k + khi) * NOUT + col0 + m];
    b.y = Wr[(k + khi + 1) * NOUT + col0 + m];
#if __HIP_DEVICE_COMPILE__
    acc = __builtin_amdgcn_wmma_f32_16x16x4_f32(false, a, false, b,
                                                (short)0, acc, false, false);
#endif
  }

  // epilogue: bias (+ BN eval + ReLU), column fixed per lane
  const int c  = col0 + m;
  const float bv = bias[c];
  float g = 0.f, bt = 0.f, mu = 0.f, is = 0.f;
  if (BNRELU) {
    g  = gamma[c];
    bt = beta[c];
    mu = mean[c];
    is = rsqrtf(var[c] + BN_EPS);
  }
  const int mbase = row0 + ((lane >> 4) << 3);  // lanes 16-31 -> rows 8..15
#pragma unroll
  for (int r = 0; r < 8; ++r) {
    float v = acc[r] + bv;
    if (BNRELU) {
      v = (v - mu) * (g * is) + bt;
      v = fmaxf(v, 0.0f);
    }
    H[(long long)(mbase + r) * NOUT + c] = v;
  }
}

extern "C" void kernel_launch(void* const* d_in, const int* in_sizes, int n_in,
                              void* d_out, int out_size, void* d_ws, size_t ws_size,
                              hipStream_t stream) {
  const float* x = (const float*)d_in[0];
  const long long* ei = (const long long*)d_in[1];
  const float* Wl0 = (const float*)d_in[2];
  const float* bl0 = (const float*)d_in[3];
  const float* Wr0 = (const float*)d_in[4];
  const float* g0 = (const float*)d_in[5];
  const float* be0 = (const float*)d_in[6];
  const float* m0 = (const float*)d_in[7];
  const float* v0 = (const float*)d_in[8];
  const float* Wl1 = (const float*)d_in[9];
  const float* bl1 = (const float*)d_in[10];
  const float* Wr1 = (const float*)d_in[11];
  const float* g1 = (const float*)d_in[12];
  const float* be1 = (const float*)d_in[13];
  const float* m1 = (const float*)d_in[14];
  const float* v1 = (const float*)d_in[15];
  const float* Wl2 = (const float*)d_in[16];
  const float* bl2 = (const float*)d_in[17];
  const float* Wr2 = (const float*)d_in[18];

  const int E = in_sizes[1] / 2;
  const long long* src = ei;
  const long long* dst = ei + E;

  // workspace layout
  char* ws = (char*)d_ws;
  const size_t feat_bytes = (size_t)NN * FDIM * sizeof(float);
  float* inv_cnt = (float*)ws;                      ws += ((size_t)NN * 4 + 511) & ~511ull;
  float* agg = (float*)ws;                          ws += feat_bytes;
  float* h0  = (float*)ws;                          ws += feat_bytes;
  float* h1  = (float*)ws;

  // degrees -> reciprocals (shared by all layers)
  hipMemsetAsync(inv_cnt, 0, (size_t)NN * sizeof(float), stream);
  count_edges<<<(E + 255) / 256, 256, 0, stream>>>(dst, inv_cnt, E);
  invert_counts<<<(NN + 255) / 256, 256, 0, stream>>>(inv_cnt, NN);

  const long long sc_threads = (long long)E * 32;
  const int sc_grid = (int)((sc_threads + 255) / 256);
  const int gm_grid = NN / 16;  // 50000 = 3125 * 16 exactly

  // layer 0
  hipMemsetAsync(agg, 0, feat_bytes, stream);
  scatter_add_f128<<<sc_grid, 256, 0, stream>>>(x, src, dst, agg, E);
  sage_gemm<128, true><<<gm_grid, 256, 0, stream>>>(
      agg, x, inv_cnt, Wl0, Wr0, bl0, g0, be0, m0, v0, h0);

  // layer 1
  hipMemsetAsync(agg, 0, feat_bytes, stream);
  scatter_add_f128<<<sc_grid, 256, 0, stream>>>(h0, src, dst, agg, E);
  sage_gemm<128, true><<<gm_grid, 256, 0, stream>>>(
      agg, h0, inv_cnt, Wl1, Wr1, bl1, g1, be1, m1, v1, h1);

  // layer 2 (no BN / ReLU, NOUT = 64) -> d_out
  hipMemsetAsync(agg, 0, feat_bytes, stream);
  scatter_add_f128<<<sc_grid, 256, 0, stream>>>(h1, src, dst, agg, E);
  sage_gemm<64, false><<<gm_grid, 128, 0, stream>>>(
      agg, h1, inv_cnt, Wl2, Wr2, bl2, nullptr, nullptr, nullptr, nullptr,
      (float*)d_out);
}